// CompetingRiskTabMLoss_14139032339229
// MI455X (gfx1250) — compile-verified
//
#include <hip/hip_runtime.h>

#define KC    4
#define NCLS  5
#define MH    8
#define NBIN  65536
#define DEPS  1e-8

typedef __attribute__((ext_vector_type(2))) float v2f;
typedef __attribute__((ext_vector_type(8))) float v8f;

// ---------------- zero workspace prefix ----------------
__global__ void zero_kernel(uint4* p, int n16) {
  int i = blockIdx.x * blockDim.x + threadIdx.x;
  if (i < n16) p[i] = make_uint4(0u, 0u, 0u, 0u);
}

// ---------------- main streaming kernel ----------------
// 128 samples per block. WMMA f32 16x16x4 computes the head-mean:
// rows = 16 samples, K = 4 heads (two chained WMMAs cover 8 heads), B = 1/8.
__global__ void __launch_bounds__(128)
k1_main(const float* __restrict__ log_h, const float* __restrict__ logits,
        const float* __restrict__ dur,   const int* __restrict__ lab,
        float* __restrict__ e_arr, float* __restrict__ binS,
        unsigned* __restrict__ cnt, double* __restrict__ acc, int N)
{
  __shared__ float  raw[128 * 36];     // 128 samples x 32 floats, padded to 36
  __shared__ float  etaT[8][4][16];    // [tile][cause][sample]
  __shared__ double ceS;

  const int tid = threadIdx.x;
  if (tid == 0) ceS = 0.0;

  const long long tileBase = (long long)blockIdx.x * 128;

  // ---- stage log_h tile into LDS (coalesced b128) ----
#pragma unroll
  for (int j = 0; j < 8; ++j) {
    int l4 = tid + j * 128;        // float4 index 0..1023
    int l  = l4 << 2;              // float index
    int sl = l >> 5;               // local sample 0..127
    int o  = l & 31;               // offset within sample (mult of 4)
    long long g = tileBase + sl; if (g >= N) g = N - 1;
    float4 v = *(const float4*)(log_h + g * 32 + o);
    *(float4*)&raw[sl * 36 + o] = v;
  }
  __syncthreads();

  // ---- WMMA head-mean: 2 tiles of 16 samples per wave ----
  const int lane = tid & 31;
  const int wv   = tid >> 5;
  const int s    = lane & 15;
  const int hb   = (lane < 16) ? 0 : 2;   // heads supplied by this lane half
  const v2f bv   = {0.125f, 0.125f};      // B matrix = all 1/8

#pragma unroll
  for (int it = 0; it < 2; ++it) {
    const int t = wv * 2 + it;
    const float* rp = &raw[(t * 16 + s) * 36];
#pragma unroll
    for (int c = 0; c < 4; ++c) {
      v2f a1 = { rp[(hb + 0) * 4 + c], rp[(hb + 1) * 4 + c] };   // heads 0..3
      v2f a2 = { rp[(hb + 4) * 4 + c], rp[(hb + 5) * 4 + c] };   // heads 4..7
      v8f d  = {0.f,0.f,0.f,0.f,0.f,0.f,0.f,0.f};
      d = __builtin_amdgcn_wmma_f32_16x16x4_f32(false, a1, false, bv, (short)0, d, false, false);
      d = __builtin_amdgcn_wmma_f32_16x16x4_f32(false, a2, false, bv, (short)0, d, false, false);
      if ((lane & 15) == 0) {                 // lanes 0 and 16 hold rows 0-7 / 8-15
        float* q = &etaT[t][c][(lane >> 4) * 8];
#pragma unroll
        for (int r = 0; r < 8; ++r) q[r] = d[r];
      }
    }
  }
  __syncthreads();

  // ---- per-sample epilogue ----
  long long g = tileBase + tid;
  double cepart = 0.0;
  if (g < N) {
    float e4[4];
#pragma unroll
    for (int c = 0; c < 4; ++c) {
      float eta = etaT[tid >> 4][c][tid & 15];
      eta = fminf(50.f, fmaxf(-50.f, eta));
      e4[c] = expf(eta);
    }
    *(float4*)(e_arr + g * 4) = make_float4(e4[0], e4[1], e4[2], e4[3]);

    float dv = dur[g];
    int bin = (int)(dv * 65536.0f);
    bin = min(max(bin, 0), NBIN - 1);
#pragma unroll
    for (int c = 0; c < 4; ++c) atomicAdd(&binS[(size_t)bin * 4 + c], e4[c]);
    atomicAdd(&cnt[bin], 1u);

    // cross-entropy of head-mean logits
    const float* L = logits + g * 40;
    float m5[5] = {0.f,0.f,0.f,0.f,0.f};
#pragma unroll
    for (int m = 0; m < 8; ++m)
#pragma unroll
      for (int c2 = 0; c2 < 5; ++c2) m5[c2] += L[m * 5 + c2];
#pragma unroll
    for (int c2 = 0; c2 < 5; ++c2) m5[c2] *= 0.125f;
    float mx = m5[0];
#pragma unroll
    for (int c2 = 1; c2 < 5; ++c2) mx = fmaxf(mx, m5[c2]);
    float se = 0.f;
#pragma unroll
    for (int c2 = 0; c2 < 5; ++c2) se += expf(m5[c2] - mx);
    float lse = mx + logf(se);
    int lb = lab[g];
    cepart = (double)(lse - m5[lb]);
  }
  atomicAdd(&ceS, cepart);
  __syncthreads();
  if (tid == 0) atomicAdd(&acc[8], ceS);
}

// ---------------- per-chunk totals (256 bins per chunk) ----------------
__global__ void s1_chunk(const float* __restrict__ binS, const unsigned* __restrict__ cnt,
                         double* __restrict__ chS, unsigned* __restrict__ chC)
{
  __shared__ double sS[4];
  __shared__ unsigned sC;
  int t = threadIdx.x;
  size_t b = (size_t)blockIdx.x * 256 + t;
  if (t == 0) { sS[0] = sS[1] = sS[2] = sS[3] = 0.0; sC = 0u; }
  __syncthreads();
  float4 v = *(const float4*)(binS + b * 4);
  unsigned c = cnt[b];
  atomicAdd(&sS[0], (double)v.x);
  atomicAdd(&sS[1], (double)v.y);
  atomicAdd(&sS[2], (double)v.z);
  atomicAdd(&sS[3], (double)v.w);
  atomicAdd(&sC, c);
  __syncthreads();
  if (t == 0) {
#pragma unroll
    for (int k = 0; k < 4; ++k) chS[(size_t)blockIdx.x * 4 + k] = sS[k];
    chC[blockIdx.x] = sC;
  }
}

// ---------------- scan chunk totals (1 block, 256 threads) ----------------
// exclusive SUFFIX for sums, exclusive PREFIX for counts (Hillis-Steele)
__global__ void s2_scan(const double* __restrict__ chS, const unsigned* __restrict__ chC,
                        double* __restrict__ chTS, unsigned* __restrict__ chOff)
{
  __shared__ double ss[4][256];
  __shared__ unsigned cc[256];
  int t = threadIdx.x;
  int r = 255 - t;                              // reversed load for suffix
#pragma unroll
  for (int k = 0; k < 4; ++k) ss[k][t] = chS[(size_t)r * 4 + k];
  cc[t] = chC[t];
  __syncthreads();
  for (int off = 1; off < 256; off <<= 1) {
    double v[4]; unsigned u = 0u;
    if (t >= off) {
#pragma unroll
      for (int k = 0; k < 4; ++k) v[k] = ss[k][t - off];
      u = cc[t - off];
    }
    __syncthreads();
    if (t >= off) {
#pragma unroll
      for (int k = 0; k < 4; ++k) ss[k][t] += v[k];
      cc[t] += u;
    }
    __syncthreads();
  }
  double oS[4]; unsigned oC;
#pragma unroll
  for (int k = 0; k < 4; ++k) oS[k] = (t > 0) ? ss[k][t - 1] : 0.0;
  oC = (t > 0) ? cc[t - 1] : 0u;
#pragma unroll
  for (int k = 0; k < 4; ++k) chTS[(size_t)(255 - t) * 4 + k] = oS[k];
  chOff[t] = oC;
}

// ---------------- within-chunk scans -> T (f64 suffix) and bin offsets ----------------
__global__ void s3_apply(const float* __restrict__ binS, const unsigned* __restrict__ cnt,
                         const double* __restrict__ chTS, const unsigned* __restrict__ chOff,
                         double* __restrict__ T, unsigned* __restrict__ offs)
{
  __shared__ float  inS[4 * 256];
  __shared__ unsigned inC[256];
  __shared__ double outT[4 * 256];
  __shared__ unsigned outO[256];
  int t = threadIdx.x;
  int gch = blockIdx.x;
  size_t b = (size_t)gch * 256 + t;
  float4 v = *(const float4*)(binS + b * 4);
  inS[t * 4 + 0] = v.x; inS[t * 4 + 1] = v.y; inS[t * 4 + 2] = v.z; inS[t * 4 + 3] = v.w;
  inC[t] = cnt[b];
  __syncthreads();
  if (t == 0) {
    double run[4];
#pragma unroll
    for (int k = 0; k < 4; ++k) run[k] = chTS[(size_t)gch * 4 + k];
    for (int i = 255; i >= 0; --i)
#pragma unroll
      for (int k = 0; k < 4; ++k) { outT[i * 4 + k] = run[k]; run[k] += (double)inS[i * 4 + k]; }
    unsigned rc = chOff[gch];
    for (int i = 0; i < 256; ++i) { outO[i] = rc; rc += inC[i]; }
  }
  __syncthreads();
#pragma unroll
  for (int k = 0; k < 4; ++k) T[b * 4 + k] = outT[t * 4 + k];
  offs[b] = outO[t];
}

// ---------------- counting-sort scatter ----------------
__global__ void k3a_scatter(const float* __restrict__ dur, const int* __restrict__ evt,
                            const float* __restrict__ e_arr, const unsigned* __restrict__ offs,
                            unsigned* __restrict__ cur, float* __restrict__ d_scat,
                            float* __restrict__ e_scat, int* __restrict__ evt_scat, int N)
{
  int i = blockIdx.x * 256 + threadIdx.x;
  if (i >= N) return;
  float d = dur[i];
  int bin = min(max((int)(d * 65536.0f), 0), NBIN - 1);
  unsigned slot = offs[bin] + atomicAdd(&cur[bin], 1u);
  d_scat[slot] = d;
  *(float4*)(e_scat + (size_t)slot * 4) = *(const float4*)(e_arr + (size_t)i * 4);
  evt_scat[slot] = evt[i];
}

// ---------------- event pass: denom = T[bin] + within-bin partial ----------------
__global__ void k3b_events(const float* __restrict__ d_scat, const float* __restrict__ e_scat,
                           const int* __restrict__ evt_scat, const unsigned* __restrict__ offs,
                           const unsigned* __restrict__ cnt, const double* __restrict__ T,
                           double* __restrict__ acc, int N)
{
  __shared__ double nm[4];
  __shared__ int    nv[4];
  int t = threadIdx.x;
  if (t < 4) { nm[t] = 0.0; nv[t] = 0; }
  __syncthreads();
  int sIdx = blockIdx.x * 256 + t;
  if (sIdx < N) {
    int ev = evt_scat[sIdx];
    if (ev > 0) {
      int c = ev - 1;
      float di = d_scat[sIdx];
      int bin = min(max((int)(di * 65536.0f), 0), NBIN - 1);
      unsigned st = offs[bin], n = cnt[bin];
      __builtin_prefetch(d_scat + st, 0, 0);      // global_prefetch_b8
      __builtin_prefetch(e_scat + (size_t)st * 4, 0, 0);
      double a = 0.0;
      for (unsigned j = st; j < st + n; ++j) {
        float dj = d_scat[j];
        if (dj >= di) a += (double)e_scat[(size_t)j * 4 + c];
      }
      double denom = T[(size_t)bin * 4 + c] + a;
      double eta   = log((double)e_scat[(size_t)sIdx * 4 + c]);  // recover clipped eta
      double term  = eta - log(denom + DEPS);
      atomicAdd(&nm[c], term);
      atomicAdd(&nv[c], 1);
    }
  }
  __syncthreads();
  if (t < 4) {
    atomicAdd(&acc[t], nm[t]);
    atomicAdd(&acc[4 + t], (double)nv[t]);
  }
}

// ---------------- final combine ----------------
__global__ void k4_final(const double* __restrict__ acc, float* __restrict__ out, int N)
{
  if (threadIdx.x == 0 && blockIdx.x == 0) {
    double ls = 0.0;
    for (int c = 0; c < 4; ++c) ls += -(acc[c]) / (acc[4 + c] + DEPS);
    double ce = acc[8] / (double)N;
    out[0] = (float)(0.4 * ls + 0.6 * ce);
  }
}

extern "C" void kernel_launch(void* const* d_in, const int* in_sizes, int n_in,
                              void* d_out, int out_size, void* d_ws, size_t ws_size,
                              hipStream_t stream)
{
  const float* log_h  = (const float*)d_in[0];
  const float* logits = (const float*)d_in[1];
  const float* dur    = (const float*)d_in[2];
  const int*   evt    = (const int*)d_in[3];
  const int*   lab    = (const int*)d_in[4];
  const int N = in_sizes[2];

  char* ws = (char*)d_ws;
  size_t o = 0;
  auto nxt = [&](size_t sz) { size_t cur = o; o = (o + sz + 255) & ~(size_t)255; return cur; };
  size_t OFF_ACC  = nxt(16 * 8);
  size_t OFF_BINS = nxt((size_t)NBIN * 4 * 4);   // f32 bin sums
  size_t OFF_CNT  = nxt((size_t)NBIN * 4);
  size_t OFF_CUR  = nxt((size_t)NBIN * 4);
  size_t OFF_CHS  = nxt(256 * 4 * 8);
  size_t OFF_CHC  = nxt(256 * 4);
  size_t ZERO_END = o;
  size_t OFF_T    = nxt((size_t)NBIN * 4 * 8);   // f64 suffix sums
  size_t OFF_OFFS = nxt((size_t)NBIN * 4);
  size_t OFF_CHTS = nxt(256 * 4 * 8);
  size_t OFF_CHOF = nxt(256 * 4);
  size_t OFF_EARR = nxt((size_t)N * 4 * 4);
  size_t OFF_DSC  = nxt((size_t)N * 4);
  size_t OFF_ESC  = nxt((size_t)N * 4 * 4);
  size_t OFF_EVT  = nxt((size_t)N * 4);
  (void)OFF_EVT; (void)ws_size; (void)n_in; (void)out_size;

  double*   acc   = (double*)(ws + OFF_ACC);
  float*    binS  = (float*)(ws + OFF_BINS);
  unsigned* cnt   = (unsigned*)(ws + OFF_CNT);
  unsigned* cur   = (unsigned*)(ws + OFF_CUR);
  double*   chS   = (double*)(ws + OFF_CHS);
  unsigned* chC   = (unsigned*)(ws + OFF_CHC);
  double*   T     = (double*)(ws + OFF_T);
  unsigned* offs  = (unsigned*)(ws + OFF_OFFS);
  double*   chTS  = (double*)(ws + OFF_CHTS);
  unsigned* chOff = (unsigned*)(ws + OFF_CHOF);
  float*    e_arr = (float*)(ws + OFF_EARR);
  float*    d_sc  = (float*)(ws + OFF_DSC);
  float*    e_sc  = (float*)(ws + OFF_ESC);
  int*      ev_sc = (int*)(ws + OFF_EVT);

  int n16 = (int)(ZERO_END / 16);
  zero_kernel<<<(n16 + 255) / 256, 256, 0, stream>>>((uint4*)ws, n16);

  int nTiles = (N + 127) / 128;
  k1_main<<<nTiles, 128, 0, stream>>>(log_h, logits, dur, lab, e_arr, binS, cnt, acc, N);

  s1_chunk<<<256, 256, 0, stream>>>(binS, cnt, chS, chC);
  s2_scan<<<1, 256, 0, stream>>>(chS, chC, chTS, chOff);
  s3_apply<<<256, 256, 0, stream>>>(binS, cnt, chTS, chOff, T, offs);

  int nB = (N + 255) / 256;
  k3a_scatter<<<nB, 256, 0, stream>>>(dur, evt, e_arr, offs, cur, d_sc, e_sc, ev_sc, N);
  k3b_events<<<nB, 256, 0, stream>>>(d_sc, e_sc, ev_sc, offs, cnt, T, acc, N);
  k4_final<<<1, 64, 0, stream>>>(acc, (float*)d_out, N);
}